// TransformLayer_43516608643464
// MI455X (gfx1250) — compile-verified
//
#include <hip/hip_runtime.h>

namespace {

constexpr int kN = 16384;
constexpr int kE = 9 * kN;
constexpr int kC = 512;

typedef __attribute__((ext_vector_type(2))) float v2f;
typedef __attribute__((ext_vector_type(8))) float v8f;

__device__ __forceinline__ v8f wmma4(v2f a, v2f b, v8f c) {
  // D = A(16x4 f32) * B(4x16 f32) + C(16x16 f32)
  return __builtin_amdgcn_wmma_f32_16x16x4_f32(false, a, false, b, (short)0, c,
                                               false, false);
}

__global__ void col_mean_kernel(const float* __restrict__ X, float* __restrict__ mean,
                                int nrows, int ncols) {
  int c = blockIdx.x * blockDim.x + threadIdx.x;
  if (c >= ncols) return;
  float s = 0.f;
  for (int n = 0; n < nrows; ++n) s += X[(size_t)n * ncols + c];
  mean[c] = s / (float)nrows;
}

__global__ void center_kernel(const float* __restrict__ X, const float* __restrict__ mean,
                              float* __restrict__ Y, int total, int cmask) {
  int i = blockIdx.x * blockDim.x + threadIdx.x;
  if (i >= total) return;
  Y[i] = X[i] - mean[i & cmask];
}

__global__ void fill_bias_kernel(float* __restrict__ out, const float* __restrict__ b,
                                 const float* __restrict__ add, int total, int cmask) {
  int i = blockIdx.x * blockDim.x + threadIdx.x;
  if (i >= total) return;
  float v = b ? b[i & cmask] : 0.f;
  if (add) v += add[i & cmask];
  out[i] = v;
}

__global__ void relu_kernel(float* __restrict__ x, int total) {
  int i = blockIdx.x * blockDim.x + threadIdx.x;
  if (i >= total) return;
  float v = x[i];
  x[i] = v > 0.f ? v : 0.f;
}

// Out[M x Co] = X[M x K] * W[K x Co], row-major f32.
// Wave computes a 64x32 tile: 4 row-subtiles x 2 col-subtiles of 16x16,
// K-loop software-pipelined (register double buffering) so fragment loads
// for step k+4 are in flight while the 8 WMMAs for step k execute.
// Block = 8 waves = 512 rows. grid = (M/512, Co/32). K % 4 == 0.
__global__ void gemm_wmma_kernel(const float* __restrict__ X, const float* __restrict__ W,
                                 float* __restrict__ Out, int K, int Co) {
  const int lane = threadIdx.x & 31;
  const int wave = threadIdx.x >> 5;
  const int r = lane & 15;     // row/col within fragment
  const int hi = lane >> 4;    // lane-half selects K+0/1 vs K+2/3
  const int koff = hi * 2;
  const int m0 = (blockIdx.x * 8 + wave) * 64;
  const int n0 = blockIdx.y * 32;

  const v8f vzero = {};
  v8f acc[4][2];
#pragma unroll
  for (int t = 0; t < 4; ++t)
#pragma unroll
    for (int u = 0; u < 2; ++u) acc[t][u] = vzero;

  const float* xp[4];
#pragma unroll
  for (int t = 0; t < 4; ++t) xp[t] = X + (size_t)(m0 + t * 16 + r) * K + koff;
  const float* wp = W + (size_t)koff * Co + n0 + r;

  // Prologue: fragments for k = 0
  v2f a[4], b[2];
#pragma unroll
  for (int t = 0; t < 4; ++t) a[t] = *(const v2f*)(xp[t]);
#pragma unroll
  for (int u = 0; u < 2; ++u) { b[u].x = wp[u * 16]; b[u].y = wp[Co + u * 16]; }

  for (int k = 4; k < K; k += 4) {
    // Issue next iteration's loads first (overlap with WMMAs below)
    v2f an[4], bn[2];
    const float* wpn = wp + (size_t)4 * Co;
#pragma unroll
    for (int t = 0; t < 4; ++t) an[t] = *(const v2f*)(xp[t] + k);
#pragma unroll
    for (int u = 0; u < 2; ++u) { bn[u].x = wpn[u * 16]; bn[u].y = wpn[Co + u * 16]; }
#pragma unroll
    for (int t = 0; t < 4; ++t)
#pragma unroll
      for (int u = 0; u < 2; ++u) acc[t][u] = wmma4(a[t], b[u], acc[t][u]);
#pragma unroll
    for (int t = 0; t < 4; ++t) a[t] = an[t];
    b[0] = bn[0]; b[1] = bn[1];
    wp = wpn;
  }
#pragma unroll
  for (int t = 0; t < 4; ++t)
#pragma unroll
    for (int u = 0; u < 2; ++u) acc[t][u] = wmma4(a[t], b[u], acc[t][u]);

#pragma unroll
  for (int v = 0; v < 8; ++v) {
    int row = v + hi * 8;   // C/D layout: lanes 16-31 hold M = v+8
#pragma unroll
    for (int t = 0; t < 4; ++t)
#pragma unroll
      for (int u = 0; u < 2; ++u)
        Out[(size_t)(m0 + t * 16 + row) * Co + n0 + u * 16 + r] = acc[t][u][v];
  }
}

// acc[dst[e]*Co + c] += Z[src[e]*Co + c] for edges with sel[e] == s
__global__ void scatter_add_kernel(float* __restrict__ acc, const float* __restrict__ Z,
                                   const int* __restrict__ src, const int* __restrict__ dst,
                                   const int* __restrict__ sel, int s, long total,
                                   int coShift) {
  long i = (long)blockIdx.x * blockDim.x + threadIdx.x;
  if (i >= total) return;
  int e = (int)(i >> coShift);
  int c = (int)(i & ((1 << coShift) - 1));
  if (sel[e] != s) return;
  atomicAdd(acc + (((size_t)dst[e]) << coShift) + c,
            Z[(((size_t)src[e]) << coShift) + c]);
}

// G += invN * H^T H ; H: [nrows x 32], G: [32 x 32] pre-zeroed. K split over waves.
__global__ void gram_wmma_kernel(const float* __restrict__ H, float* __restrict__ G,
                                 int nrows, float invN) {
  const int lane = threadIdx.x & 31;
  const int wave = threadIdx.x >> 5;
  const int r = lane & 15;
  const int hi = lane >> 4;
  const int koff = hi * 2;
  const int nw = gridDim.x * 8;
  const int slice = nrows / nw;
  const int kbase = (blockIdx.x * 8 + wave) * slice;
  v8f a00 = {}, a01 = {}, a10 = {}, a11 = {};
  for (int k = 0; k < slice; k += 4) {
    int kk = kbase + k + koff;
    v2f f0, f1;                       // A tile == B tile (symmetric gram)
    f0.x = H[(size_t)kk * 32 + r];
    f0.y = H[(size_t)(kk + 1) * 32 + r];
    f1.x = H[(size_t)kk * 32 + 16 + r];
    f1.y = H[(size_t)(kk + 1) * 32 + 16 + r];
    a00 = wmma4(f0, f0, a00);
    a01 = wmma4(f0, f1, a01);
    a10 = wmma4(f1, f0, a10);
    a11 = wmma4(f1, f1, a11);
  }
#pragma unroll
  for (int v = 0; v < 8; ++v) {
    int row = v + hi * 8;
    atomicAdd(&G[row * 32 + r],             a00[v] * invN);
    atomicAdd(&G[row * 32 + 16 + r],        a01[v] * invN);
    atomicAdd(&G[(16 + row) * 32 + r],      a10[v] * invN);
    atomicAdd(&G[(16 + row) * 32 + 16 + r], a11[v] * invN);
  }
}

// out[i] = fcb[i] + sum_j fcW[i,j] * g[j]   (torch Linear: g @ W^T + b)
__global__ void fc1024_kernel(const float* __restrict__ W, const float* __restrict__ b,
                              const float* __restrict__ g, float* __restrict__ out) {
  int i = blockIdx.x * blockDim.x + threadIdx.x;
  if (i >= 1024) return;
  float s = b[i];
  for (int j = 0; j < 1024; ++j) s = fmaf(W[(size_t)i * 1024 + j], g[j], s);
  out[i] = s;
}

__global__ void matmul32_kernel(const float* __restrict__ A, const float* __restrict__ B,
                                float* __restrict__ Cm) {
  int j = threadIdx.x & 31;
  int i = threadIdx.x >> 5;
  float s = 0.f;
  for (int k = 0; k < 32; ++k) s = fmaf(A[i * 32 + k], B[k * 32 + j], s);
  Cm[i * 32 + j] = s;
}

__global__ void transpose32_kernel(const float* __restrict__ in, float* __restrict__ out) {
  int i = threadIdx.x >> 5;
  int j = threadIdx.x & 31;
  out[j * 32 + i] = in[i * 32 + j];
}

inline int ilog2(int v) { int s = 0; while ((1 << s) < v) ++s; return s; }

inline void launch_gemm(const float* X, const float* W, float* Out, int Mrows, int K,
                        int Co, hipStream_t stream) {
  dim3 grid(Mrows / 512, Co / 32);
  gemm_wmma_kernel<<<grid, 256, 0, stream>>>(X, W, Out, K, Co);
}

inline void launch_scatter(float* acc, const float* Z, const int* src, const int* dst,
                           const int* sel, int s, int coShift, hipStream_t stream) {
  long total = (long)kE << coShift;
  int blocks = (int)((total + 255) / 256);
  scatter_add_kernel<<<blocks, 256, 0, stream>>>(acc, Z, src, dst, sel, s, total, coShift);
}

inline void launch_fill(float* out, const float* b, const float* add, int total,
                        int cmask, hipStream_t stream) {
  fill_bias_kernel<<<(total + 255) / 256, 256, 0, stream>>>(out, b, add, total, cmask);
}

// One SelectionConv layer: acc = bias; for s: Z = X @ W[s]; acc[dst]+=Z[src] (sel==s)
inline void sel_layer(const float* Xin, const float* W, const float* b, int K, int Co,
                      const int* src, const int* dst, const int* sel, int nsel,
                      float* accOut, float* ZB, bool relu, hipStream_t stream) {
  int cmask = Co - 1;
  int coShift = ilog2(Co);
  launch_fill(accOut, b, nullptr, kN * Co, cmask, stream);
  for (int s = 0; s < nsel; ++s) {
    launch_gemm(Xin, W + (size_t)s * K * Co, ZB, kN, K, Co, stream);
    launch_scatter(accOut, ZB, src, dst, sel, s, coShift, stream);
  }
  if (relu) relu_kernel<<<(kN * Co + 255) / 256, 256, 0, stream>>>(accOut, kN * Co);
}

inline void run_net(const float* X0, const int* src, const int* dst, const int* sel,
                    const float* W1, const float* b1, const float* W2, const float* b2,
                    const float* W3, const float* b3, const float* fcW, const float* fcb,
                    float* H1, float* H2, float* H3, float* ZB, float* GRAM,
                    float* OUTMAT, hipStream_t stream) {
  sel_layer(X0, W1, b1, 512, 256, src, dst, sel, 9, H1, ZB, true, stream);
  sel_layer(H1, W2, b2, 256, 128, src, dst, sel, 9, H2, ZB, true, stream);
  sel_layer(H2, W3, b3, 128, 32, src, dst, sel, 9, H3, ZB, false, stream);
  launch_fill(GRAM, nullptr, nullptr, 1024, 0, stream);
  gram_wmma_kernel<<<4, 256, 0, stream>>>(H3, GRAM, kN, 1.0f / (float)kN);
  fc1024_kernel<<<4, 256, 0, stream>>>(fcW, fcb, GRAM, OUTMAT);
}

}  // namespace

extern "C" void kernel_launch(void* const* d_in, const int* in_sizes, int n_in,
                              void* d_out, int out_size, void* d_ws, size_t ws_size,
                              hipStream_t stream) {
  (void)in_sizes; (void)n_in; (void)out_size; (void)ws_size;
  const float* cF   = (const float*)d_in[0];
  const float* sF   = (const float*)d_in[1];
  const int*   cEI  = (const int*)d_in[2];
  const int*   cSel = (const int*)d_in[3];
  const int*   sEI  = (const int*)d_in[4];
  const int*   sSel = (const int*)d_in[5];
  const float* compW = (const float*)d_in[6];
  const float* compB = (const float*)d_in[7];
  const float* unzW  = (const float*)d_in[8];
  const float* unzB  = (const float*)d_in[9];
  const float* cW1 = (const float*)d_in[10]; const float* cB1 = (const float*)d_in[11];
  const float* cW2 = (const float*)d_in[12]; const float* cB2 = (const float*)d_in[13];
  const float* cW3 = (const float*)d_in[14]; const float* cB3 = (const float*)d_in[15];
  const float* cFcW = (const float*)d_in[16]; const float* cFcB = (const float*)d_in[17];
  const float* sW1 = (const float*)d_in[18]; const float* sB1 = (const float*)d_in[19];
  const float* sW2 = (const float*)d_in[20]; const float* sB2 = (const float*)d_in[21];
  const float* sW3 = (const float*)d_in[22]; const float* sB3 = (const float*)d_in[23];
  const float* sFcW = (const float*)d_in[24]; const float* sFcB = (const float*)d_in[25];

  const int* cSrc = cEI;       const int* cDst = cEI + kE;
  const int* sSrc = sEI;       const int* sDst = sEI + kE;

  float* p = (float*)d_ws;
  auto wsalloc = [&](size_t n) { float* q = p; p += n; return q; };
  float* XC    = wsalloc((size_t)kN * kC);    // centered content features
  float* XS    = wsalloc((size_t)kN * kC);    // centered style features
  float* H1    = wsalloc((size_t)kN * 256);
  float* H2    = wsalloc((size_t)kN * 128);
  float* H3    = wsalloc((size_t)kN * 32);
  float* ZB    = wsalloc((size_t)kN * 256);   // per-selection GEMM scratch
  float* CC    = wsalloc((size_t)kN * 32);    // compress_content
  float* TF    = wsalloc((size_t)kN * 32);    // transfeature
  float* CMEAN = wsalloc(512);
  float* SMEAN = wsalloc(512);
  float* GRAM  = wsalloc(1024);
  float* CMAT  = wsalloc(1024);
  float* SMAT  = wsalloc(1024);
  float* TMAT  = wsalloc(1024);
  float* TMATT = wsalloc(1024);

  // Column means and centering
  col_mean_kernel<<<2, 256, 0, stream>>>(cF, CMEAN, kN, kC);
  col_mean_kernel<<<2, 256, 0, stream>>>(sF, SMEAN, kN, kC);
  center_kernel<<<(kN * kC) / 256, 256, 0, stream>>>(cF, CMEAN, XC, kN * kC, kC - 1);
  center_kernel<<<(kN * kC) / 256, 256, 0, stream>>>(sF, SMEAN, XS, kN * kC, kC - 1);

  // compress_content: kernel_size=1 -> only sel==0 edges contribute
  sel_layer(XC, compW, compB, 512, 32, cSrc, cDst, cSel, 1, CC, ZB, false, stream);

  // cMat / sMat
  run_net(XC, cSrc, cDst, cSel, cW1, cB1, cW2, cB2, cW3, cB3, cFcW, cFcB,
          H1, H2, H3, ZB, GRAM, CMAT, stream);
  run_net(XS, sSrc, sDst, sSel, sW1, sB1, sW2, sB2, sW3, sB3, sFcW, sFcB,
          H1, H2, H3, ZB, GRAM, SMAT, stream);

  // transmatrix = sMat @ cMat ; transfeature = CC @ transmatrix^T
  matmul32_kernel<<<1, 1024, 0, stream>>>(SMAT, CMAT, TMAT);
  transpose32_kernel<<<1, 1024, 0, stream>>>(TMAT, TMATT);
  launch_gemm(CC, TMATT, TF, kN, 32, 32, stream);

  // out = scatter_{sel==0}(TF @ unzip_W) + unzip_b + sMean
  float* out = (float*)d_out;
  launch_fill(out, unzB, SMEAN, kN * kC, kC - 1, stream);
  launch_gemm(TF, unzW, XC, kN, 32, kC, stream);  // reuse XC as unzip scratch
  launch_scatter(out, XC, cSrc, cDst, cSel, 0, ilog2(kC), stream);

  // second output: transmatrix (32x32) appended flat
  hipMemcpyAsync(out + (size_t)kN * kC, TMAT, 1024 * sizeof(float),
                 hipMemcpyDeviceToDevice, stream);
}